// ForwardOperator_19731079758152
// MI455X (gfx1250) — compile-verified
//
#include <hip/hip_runtime.h>
#include <math.h>

// Radon forward projection for MI455X (gfx1250).
// B=8 images of 256x256 f32, 360 angles, output (8,1,360,256) f32.
//
// - Image (256KB) staged into LDS by the CDNA5 Tensor Data Mover
//   (tensor_load_to_lds, TENSORcnt) with TDM row padding (pad_interval=7,
//   pad_amount=1 DWORD) -> LDS row stride 257 floats -> conflict-free banks
//   at theta ~ 90 deg. Fallback: async global->LDS b128 copies (ASYNCcnt).
// - Per (thread, angle) the sample coords are affine in h; the fully
//   in-bounds h-interval runs a lean bilinear body, boundary samples use
//   exact masked reference semantics.

#define D        256
#define NA       360
#define NB       8
#define APB      4            // angles per block
#define TPB      512          // 16 wave32 waves
#define LSTRIDE  257          // padded LDS row stride in floats

#if defined(__has_builtin)
#if __has_builtin(__builtin_amdgcn_tensor_load_to_lds)
#define HAVE_TDM 1
#endif
#endif

typedef unsigned int v4u __attribute__((ext_vector_type(4)));
typedef int          v4i __attribute__((ext_vector_type(4)));
typedef int          v8i __attribute__((ext_vector_type(8)));

__device__ __forceinline__ void sample_general(const float* t, float ix, float iy,
                                               float& acc) {
  const float x0f = floorf(ix);
  const float y0f = floorf(iy);
  const float fx  = ix - x0f;
  const float fy  = iy - y0f;
  const int x0 = (int)x0f, y0 = (int)y0f;
  const int x1 = x0 + 1,   y1 = y0 + 1;

  float wx0 = 1.0f - fx, wx1 = fx;
  float wy0 = 1.0f - fy, wy1 = fy;
  // zero weights for out-of-range corners (matches reference: OOB contributes 0)
  wx0 = ((unsigned)x0 < (unsigned)D) ? wx0 : 0.0f;
  wx1 = ((unsigned)x1 < (unsigned)D) ? wx1 : 0.0f;
  wy0 = ((unsigned)y0 < (unsigned)D) ? wy0 : 0.0f;
  wy1 = ((unsigned)y1 < (unsigned)D) ? wy1 : 0.0f;

  const int cx0 = x0 < 0 ? 0 : (x0 > D - 1 ? D - 1 : x0);
  const int cx1 = x1 < 0 ? 0 : (x1 > D - 1 ? D - 1 : x1);
  const int cy0 = y0 < 0 ? 0 : (y0 > D - 1 ? D - 1 : y0);
  const int cy1 = y1 < 0 ? 0 : (y1 > D - 1 ? D - 1 : y1);

  acc = fmaf(wx0 * wy0, t[cy0 * LSTRIDE + cx0], acc);
  acc = fmaf(wx1 * wy0, t[cy0 * LSTRIDE + cx1], acc);
  acc = fmaf(wx0 * wy1, t[cy1 * LSTRIDE + cx0], acc);
  acc = fmaf(wx1 * wy1, t[cy1 * LSTRIDE + cx1], acc);
}

__device__ __forceinline__ void sample_interior(const float* t, float ix, float iy,
                                                float& acc) {
  const float x0f = floorf(ix);
  const float y0f = floorf(iy);
  const float fx  = ix - x0f;
  const float fy  = iy - y0f;
  const int base  = (int)y0f * LSTRIDE + (int)x0f;

  const float v00 = t[base];
  const float v10 = t[base + 1];
  const float v01 = t[base + LSTRIDE];
  const float v11 = t[base + LSTRIDE + 1];

  const float wx0 = 1.0f - fx;
  const float wy0 = 1.0f - fy;
  acc = fmaf(wx0 * wy0, v00, acc);
  acc = fmaf(fx  * wy0, v10, acc);
  acc = fmaf(wx0 * fy,  v01, acc);
  acc = fmaf(fx  * fy,  v11, acc);
}

// Tighten [lo,hi] so that base + slope*h stays in [0, cmax] for h in [lo,hi].
__device__ __forceinline__ void tighten(float base, float slope, float cmax,
                                        float& lo, float& hi) {
  if (slope > 1e-6f) {
    lo = fmaxf(lo, -base / slope);
    hi = fminf(hi, (cmax - base) / slope);
  } else if (slope < -1e-6f) {
    lo = fmaxf(lo, (cmax - base) / slope);
    hi = fminf(hi, -base / slope);
  } else if (base < 0.0f || base > cmax) {
    lo = 1.0e9f; hi = -1.0e9f;
  }
}

__global__ __launch_bounds__(TPB) void radon_fwd_kernel(
    const float* __restrict__ x, float* __restrict__ out) {
  __shared__ float tile[D * LSTRIDE];   // 263,168 bytes < 320KB/WGP

  const int tid   = threadIdx.x;
  const int blk   = blockIdx.x;               // 0..719
  const int b     = blk / (NA / APB);         // batch 0..7
  const int abase = (blk % (NA / APB)) * APB; // first angle of block

  const float* img = x + (size_t)b * (D * D);

#if HAVE_TDM
  // ---- Stage image b into LDS with one Tensor Data Mover descriptor ----
  if ((tid >> 5) == 0) {   // wave 0 only (TDM ignores EXEC; avoid 16x DMA)
    const unsigned long long ga = (unsigned long long)(const void*)img;
    v4u g0;
    g0.x = 0x1u;                                   // count=1, user descriptor
    g0.y = (unsigned)(size_t)tile;                 // lds_addr (bytes)
    g0.z = (unsigned)(ga & 0xFFFFFFFFull);         // global_addr[31:0]
    g0.w = ((unsigned)(ga >> 32) & 0x01FFFFFFu)    // global_addr[56:32]
           | (2u << 30);                           // type=2 ("image")
    v8i g1;
    g1[0] = (int)((2u << 16)      // data_size = 4 bytes
                | (1u << 20)      // pad_enable
                | (7u << 22));    // pad_interval: every 256 DWORDs (one row)
                                  // pad_amount = 0 -> 1 DWORD pad => stride 257
    g1[1] = (int)(256u << 16);    // tensor_dim0 = 256 (bits 79:48, low half)
    g1[2] = (int)(256u << 16);    // tensor_dim1 = 256 (bits 111:80, low half)
    g1[3] = (int)(256u << 16);    // tile_dim0   = 256 (bits 127:112)
    g1[4] = 256;                  // tile_dim1 = 256, tile_dim2 = 0
    g1[5] = 256;                  // tensor_dim0_stride = 256 (bits 191:160)
    g1[6] = 0;                    // dim0_stride hi / dim1_stride low16 = 0
    g1[7] = 1;                    // tensor_dim1_stride = 65536 (bits 255:224)
    const v4i z4 = {0, 0, 0, 0};  // 2-D tensor: groups 2/3 unused
#if __clang_major__ >= 23
    const v8i z8 = {0, 0, 0, 0, 0, 0, 0, 0};
    __builtin_amdgcn_tensor_load_to_lds(g0, g1, z4, z4, z8, 0);
#else
    __builtin_amdgcn_tensor_load_to_lds(g0, g1, z4, z4, 0);
#endif
    __builtin_amdgcn_s_wait_tensorcnt(0);
  }
#else
  // ---- Fallback: async global->LDS b128 copies ----
  for (int i = tid; i < (D * D / 4); i += TPB) {
    const int row = i >> 6;
    const int c4  = (i & 63) << 2;
    unsigned long long gaddr =
        (unsigned long long)(const void*)(img + row * D + c4);
    unsigned lds_addr = (unsigned)(size_t)(tile + row * LSTRIDE + c4);
    asm volatile("global_load_async_to_lds_b128 %0, %1, off"
                 :: "v"(lds_addr), "v"(gaddr)
                 : "memory");
  }
  asm volatile("s_wait_asynccnt 0x0" ::: "memory");
#endif

  // zero the pad column (never read when classification is exact; safety net)
  for (int i = tid; i < D; i += TPB) tile[i * LSTRIDE + D] = 0.0f;
  __syncthreads();

  // ---- Compute projections ----
  const int   w   = tid & (D - 1);              // detector position
  const int   g   = tid >> 8;                   // 0 or 1: interleave angles
  const float xw  = (2.0f * (float)w + 1.0f) * (1.0f / (float)D) - 1.0f;
  const float yh0 = (1.0f / (float)D) - 1.0f;   // yh at h=0

  for (int aa = g; aa < APB; aa += (TPB / D)) {
    const int a = abase + aa;
    // reference: angles = linspace(0,180,361)[:-1] -> 0.5*a degrees
    const float ang = (float)a * (float)(M_PI / 360.0);
    const float sv = sinf(ang);
    const float cv = cosf(ang);

    // ix/iy at h=0; per-h step is exactly (-sv, +cv)
    const float gx0 = cv * xw - sv * yh0;
    const float gy0 = sv * xw + cv * yh0;
    const float ixb = (gx0 + 1.0f) * (0.5f * (float)D) - 0.5f;
    const float iyb = (gy0 + 1.0f) * (0.5f * (float)D) - 0.5f;

    // h-interval where the whole 2x2 footprint is in-bounds
    float lo = 0.0f, hi = 255.0f;
    tighten(ixb, -sv, 254.99f, lo, hi);
    tighten(iyb,  cv, 254.99f, lo, hi);
    int hlo = (int)ceilf(lo) + 1;    // shrink by 1 step each side: rounding in
    int hhi = (int)floorf(hi) - 1;   // the bounds can never leak OOB samples
    hlo = hlo < 0   ? 0   : (hlo > D ? D : hlo);
    hhi = hhi > D-1 ? D-1 : hhi;
    if (hlo > hhi) { hlo = D; hhi = D - 1; }  // empty interior

    float acc = 0.0f;
    int h = 0;
    for (; h < hlo; ++h) {
      const float hf = (float)h;
      sample_general(tile, fmaf(hf, -sv, ixb), fmaf(hf, cv, iyb), acc);
    }
#pragma unroll 4
    for (; h <= hhi; ++h) {
      const float hf = (float)h;
      sample_interior(tile, fmaf(hf, -sv, ixb), fmaf(hf, cv, iyb), acc);
    }
    for (; h < D; ++h) {
      const float hf = (float)h;
      sample_general(tile, fmaf(hf, -sv, ixb), fmaf(hf, cv, iyb), acc);
    }

    out[((size_t)b * NA + a) * D + w] = acc * (1.0f / (float)D);
  }
}

extern "C" void kernel_launch(void* const* d_in, const int* in_sizes, int n_in,
                              void* d_out, int out_size, void* d_ws, size_t ws_size,
                              hipStream_t stream) {
  const float* x = (const float*)d_in[0];
  float* out = (float*)d_out;
  (void)in_sizes; (void)n_in; (void)out_size; (void)d_ws; (void)ws_size;

  dim3 grid(NB * (NA / APB));   // 720 workgroups
  dim3 block(TPB);              // 512 threads = 16 wave32 waves
  radon_fwd_kernel<<<grid, block, 0, stream>>>(x, out);
}